// SelectiveSSM_12816182411349
// MI455X (gfx1250) — compile-verified
//
#include <hip/hip_runtime.h>
#include <hip/hip_bf16.h>

// ---------------------------------------------------------------------------
// Selective SSM (Mamba scan) forward for MI455X / gfx1250.
//   x (2,512,4096) f32 ; A_log (4096,16) ; D (4096) ; x_proj_w (160,4096) ;
//   dt_proj_w (4096,128) ; dt_proj_b (4096)  ->  y (2,512,4096) f32
//
// K1 : xp = x @ x_proj_w^T        split-K x4, WMMA f32 16x16x4, b128 loads
// K1r: fixed-order split-K reduce (deterministic)
// K2 : dt = softplus(xp[:, :128] @ dt_proj_w^T + b)   WMMA, b128 loads
// K3 : fused scan, one lane per (d, n) state, shfl-xor dot over n
// ---------------------------------------------------------------------------

#define BATCH   2
#define SEQ     512
#define DIN     4096
#define DSTATE  16
#define DTRANK  128
#define NPROJ   160
#define NROWS   (BATCH * SEQ)   // 1024
#define KSPLIT  4

typedef float v2f __attribute__((ext_vector_type(2)));
typedef float v4f __attribute__((ext_vector_type(4)));
typedef float v8f __attribute__((ext_vector_type(8)));

// Two WMMAs fed from one b128 per matrix.  K-slot permutation is legal as
// long as A and B use the same physical-k assignment:
//   lane hi half loads [k+4*hi .. k+4*hi+3];
//   WMMA#1 uses floats {0,1} -> physical k {0,1} (hi=0) / {4,5} (hi=1)
//   WMMA#2 uses floats {2,3} -> physical k {2,3} (hi=0) / {6,7} (hi=1)
__device__ __forceinline__ v8f wmma_pair(v8f c, v4f a, v4f b) {
  v2f a0 = {a.x, a.y}, a1 = {a.z, a.w};
  v2f b0 = {b.x, b.y}, b1 = {b.z, b.w};
  c = __builtin_amdgcn_wmma_f32_16x16x4_f32(false, a0, false, b0,
                                            (short)0, c, false, false);
  c = __builtin_amdgcn_wmma_f32_16x16x4_f32(false, a1, false, b1,
                                            (short)0, c, false, false);
  return c;
}

// ---------------------------------------------------------------------------
// K1: split-K partial GEMM.  2560 waves: (ks, mt, nt) = (4, 64, 10).
// ---------------------------------------------------------------------------
__global__ __launch_bounds__(256) void gemm_xproj_partial(
    const float* __restrict__ x, const float* __restrict__ w,
    float* __restrict__ part) {
  const int lane  = threadIdx.x & 31;
  const int wave  = threadIdx.x >> 5;
  const int gtile = blockIdx.x * 8 + wave;     // 2560
  const int ks    = gtile / 640;
  const int tile  = gtile - ks * 640;
  const int mt = tile / 10, nt = tile - mt * 10;
  const int m0 = mt * 16, n0 = nt * 16;
  const int lo = lane & 15, hi = lane >> 4;
  const int kbase = ks * (DIN / KSPLIT);       // chunk of 1024

  const float* aptr = x + (size_t)(m0 + lo) * DIN + kbase + 4 * hi;
  const float* bptr = w + (size_t)(n0 + lo) * DIN + kbase + 4 * hi;

  v8f c = {};
#pragma unroll 4
  for (int k = 0; k < DIN / KSPLIT; k += 8) {
    v4f a = *(const v4f*)(aptr + k);
    v4f b = *(const v4f*)(bptr + k);
    c = wmma_pair(c, a, b);
  }

  float* out = part + (size_t)ks * NROWS * NPROJ + n0 + lo;
  const int rbase = m0 + 8 * hi;
#pragma unroll
  for (int j = 0; j < 8; ++j)
    out[(size_t)(rbase + j) * NPROJ] = c[j];
}

// Fixed-order reduction over the 4 K-chunks (deterministic).
__global__ __launch_bounds__(256) void reduce_xp_kernel(
    const float* __restrict__ part, float* __restrict__ xp) {
  const size_t S = (size_t)NROWS * NPROJ;
  const size_t i = (size_t)blockIdx.x * 256 + threadIdx.x;   // 163840 total
  xp[i] = ((part[i] + part[i + S]) + part[i + 2 * S]) + part[i + 3 * S];
}

// ---------------------------------------------------------------------------
// K2: dt (1024 x 4096) = softplus(xp[:, :128] @ dt_proj_w^T + bias)
// 16384 waves, K = 128 -> 16 iterations of (2 b128 loads + 2 WMMAs).
// ---------------------------------------------------------------------------
__global__ __launch_bounds__(256) void gemm_dt_kernel(
    const float* __restrict__ xp, const float* __restrict__ w,
    const float* __restrict__ bias, float* __restrict__ dtb) {
  const int lane = threadIdx.x & 31;
  const int wave = threadIdx.x >> 5;
  const int tile = blockIdx.x * 8 + wave;      // 16384
  const int mt = tile >> 8;
  const int nt = tile & 255;
  const int m0 = mt * 16, n0 = nt * 16;
  const int lo = lane & 15, hi = lane >> 4;

  const float* aptr = xp + (size_t)(m0 + lo) * NPROJ + 4 * hi;
  const float* bptr = w + (size_t)(n0 + lo) * DTRANK + 4 * hi;

  v8f c = {};
#pragma unroll
  for (int k = 0; k < DTRANK; k += 8) {
    v4f a = *(const v4f*)(aptr + k);
    v4f b = *(const v4f*)(bptr + k);
    c = wmma_pair(c, a, b);
  }

  const int d = n0 + lo;
  const float bv = bias[d];
  float* out = dtb + d;
  const int rbase = m0 + 8 * hi;
#pragma unroll
  for (int j = 0; j < 8; ++j) {
    float v = c[j] + bv;
    float sp = (v > 20.0f) ? v : log1pf(__expf(v));   // softplus
    out[(size_t)(rbase + j) * DIN] = sp;
  }
}

// ---------------------------------------------------------------------------
// K3: fused scan.  One lane per (d, n) state; wave = 2 channels x 16 states.
// Block 256 covers 16 channels; grid (256, 2) -> 4096 waves on the serial
// chain.  Per step: 1 exp + 2 fma per lane, dot over n via 4 shfl_xor.
// ---------------------------------------------------------------------------
#define TC 64

__global__ __launch_bounds__(256) void scan_kernel(
    const float* __restrict__ x, const float* __restrict__ A_log,
    const float* __restrict__ Dp, const float* __restrict__ xp,
    const float* __restrict__ dtb, float* __restrict__ y) {
  __shared__ float sBC[TC * 32];   // per t: B[16] then C[16]
  __shared__ float sdt[TC * 16];   // per t: 16 channels
  __shared__ float sx [TC * 16];
  __shared__ float sy [TC * 16];

  const int tid  = threadIdx.x;
  const int lane = tid & 31;
  const int wave = tid >> 5;
  const int n    = lane & 15;          // state index
  const int grp  = lane >> 4;          // channel-within-wave
  const int c    = wave * 2 + grp;     // channel-within-block (0..15)
  const int b    = blockIdx.y;
  const int d    = blockIdx.x * 16 + c;

  const float Aval = -__expf(A_log[(size_t)d * DSTATE + n]);
  const float Dv   = Dp[d];
  float h = 0.0f;

  const float* xpb = xp  + (size_t)b * SEQ * NPROJ + DTRANK;
  const float* xg  = x   + (size_t)b * SEQ * DIN + blockIdx.x * 16;
  const float* dtg = dtb + (size_t)b * SEQ * DIN + blockIdx.x * 16;
  float*       yg  = y   + (size_t)b * SEQ * DIN + blockIdx.x * 16;

  for (int t0 = 0; t0 < SEQ; t0 += TC) {
    __syncthreads();   // previous chunk (incl. sy readback) fully done
    for (int f = tid; f < TC * 32; f += 256) {
      const int tt = f >> 5, j = f & 31;
      sBC[f] = xpb[(size_t)(t0 + tt) * NPROJ + j];
    }
    for (int f = tid; f < TC * 16; f += 256) {
      const int tt = f >> 4, j = f & 15;
      sdt[f] = dtg[(size_t)(t0 + tt) * DIN + j];
      sx [f] = xg [(size_t)(t0 + tt) * DIN + j];
    }
    __syncthreads();

    if (t0 + TC < SEQ)   // pull next B/C chunk toward the caches
      __builtin_prefetch(&xpb[(size_t)(t0 + TC) * NPROJ + (tid & 31)], 0, 0);

    for (int tl = 0; tl < TC; ++tl) {
      const float dt = sdt[tl * 16 + c];
      const float xv = sx [tl * 16 + c];
      const float Bn = sBC[tl * 32 + n];
      const float Cn = sBC[tl * 32 + 16 + n];
      h = __expf(dt * Aval) * h + (dt * xv) * Bn;
      float p = h * Cn;
      p += __shfl_xor(p, 1);
      p += __shfl_xor(p, 2);
      p += __shfl_xor(p, 4);
      p += __shfl_xor(p, 8);
      if (n == 0) sy[tl * 16 + c] = p + Dv * xv;
    }
    __syncthreads();
    for (int f = tid; f < TC * 16; f += 256) {
      const int tt = f >> 4, j = f & 15;
      yg[(size_t)(t0 + tt) * DIN + j] = sy[f];
    }
  }
}

// ---------------------------------------------------------------------------
// Host launcher.  Workspace layout (floats):
//   [0 .. 163840)                 : xp  (1024 x 160)
//   [163840 .. 163840+4194304)    : dtb (1024 x 4096)
// Split-K partials (4 x 1024 x 160 = 655360 floats) alias the dtb region;
// they are dead before K2 overwrites it.  Total ws ~16.7 MB.
// ---------------------------------------------------------------------------
extern "C" void kernel_launch(void* const* d_in, const int* in_sizes, int n_in,
                              void* d_out, int out_size, void* d_ws, size_t ws_size,
                              hipStream_t stream) {
  const float* x     = (const float*)d_in[0];
  const float* A_log = (const float*)d_in[1];
  const float* Dp    = (const float*)d_in[2];
  const float* xw    = (const float*)d_in[3];
  const float* dtw   = (const float*)d_in[4];
  const float* dtb_b = (const float*)d_in[5];
  float* y = (float*)d_out;

  float* xp   = (float*)d_ws;
  float* dtb  = xp + (size_t)NROWS * NPROJ;
  float* part = dtb;   // alias: partials live in dtb region until K2 runs

  // K1: 2560 waves (split-K x4), 8 waves/block
  gemm_xproj_partial<<<dim3(320), dim3(256), 0, stream>>>(x, xw, part);
  // K1r: deterministic fixed-order reduce, 163840 elements
  reduce_xp_kernel<<<dim3(640), dim3(256), 0, stream>>>(part, xp);
  // K2: 16384 waves
  gemm_dt_kernel<<<dim3(2048), dim3(256), 0, stream>>>(xp, dtw, dtb_b, dtb);
  // K3: one lane per (d, n); grid (DIN/16, BATCH)
  scan_kernel<<<dim3(DIN / 16, BATCH), dim3(256), 0, stream>>>(
      x, A_log, Dp, xp, dtb, y);
}